// HybridModel_87007447483118
// MI455X (gfx1250) — compile-verified
//
#include <hip/hip_runtime.h>
#include <hip/hip_bf16.h>

#define NQ      14
#define NQPAD   16      // expvals row stride in workspace (zero padded)
#define NSTATE  16384   // 2^14
#define NPAIR   8192
#define NCPAIR  4096
#define BATCH   128
#define NT      512     // threads per sim block (16 waves)

typedef float  v8f  __attribute__((ext_vector_type(8)));
typedef float  v2f  __attribute__((ext_vector_type(2)));

// ---------------- complex helpers (float2 = {re, im}) ----------------
__device__ __forceinline__ float2 cmul(float2 a, float2 b) {
    return make_float2(a.x * b.x - a.y * b.y, a.x * b.y + a.y * b.x);
}
__device__ __forceinline__ float2 cadd(float2 a, float2 b) {
    return make_float2(a.x + b.x, a.y + b.y);
}

// Apply 2x2 complex gate on bit position bp (bp = 13 - wire; wire 0 = MSB).
__device__ __forceinline__ void apply1q(float2* S, float2 g00, float2 g01,
                                        float2 g10, float2 g11, int bp, int tid) {
    const int R = 1 << bp;
    for (int p = tid; p < NPAIR; p += NT) {
        int i0 = ((p >> bp) << (bp + 1)) | (p & (R - 1));
        int i1 = i0 | R;
        float2 a0 = S[i0];
        float2 a1 = S[i1];
        S[i0] = cadd(cmul(g00, a0), cmul(g01, a1));
        S[i1] = cadd(cmul(g10, a0), cmul(g11, a1));
    }
}

// ---------------- kernel 1: statevector sim, one block per batch element ---
__global__ __launch_bounds__(NT)
void qsim_kernel(const float* __restrict__ x,
                 const float* __restrict__ qw,
                 float* __restrict__ expvals) {
    extern __shared__ float2 S[];           // 16384 complex amplitudes, 128 KB
    __shared__ float wacc[NT / 32][NQ];     // per-wave partial sums

    const int b   = blockIdx.x;
    const int tid = threadIdx.x;

    // |0...0>
    for (int i = tid; i < NSTATE; i += NT) S[i] = make_float2(0.f, 0.f);
    if (tid == 0) S[0] = make_float2(1.f, 0.f);
    __syncthreads();

    // AngleEmbedding: RX(x_w) on wire w
    for (int w = 0; w < NQ; ++w) {
        float half = 0.5f * x[b * NQ + w];
        float c = cosf(half), s = sinf(half);
        float2 g00 = make_float2(c, 0.f);
        float2 g01 = make_float2(0.f, -s);
        apply1q(S, g00, g01, g01, g00, NQ - 1 - w, tid);
        __syncthreads();
    }

    // StronglyEntanglingLayers
    for (int l = 0; l < 3; ++l) {
        for (int w = 0; w < NQ; ++w) {
            const float* g = &qw[(l * NQ + w) * 3];
            float phi = g[0], th = g[1], om = g[2];
            float ct = cosf(0.5f * th), st = sinf(0.5f * th);
            float ap = -0.5f * (phi + om), am = -0.5f * (phi - om);
            float2 ep = make_float2(cosf(ap), sinf(ap));
            float2 em = make_float2(cosf(am), sinf(am));
            float2 g00 = make_float2(ep.x * ct, ep.y * ct);
            float2 g01 = make_float2(-em.x * st, em.y * st);   // -conj(em)*st
            float2 g10 = make_float2(em.x * st, em.y * st);
            float2 g11 = make_float2(ep.x * ct, -ep.y * ct);   //  conj(ep)*ct
            apply1q(S, g00, g01, g10, g11, NQ - 1 - w, tid);
            __syncthreads();
        }
        int r = (l % (NQ - 1)) + 1;           // PennyLane default ranges: 1,2,3
        for (int w = 0; w < NQ; ++w) {
            int c = w, t = (w + r) % NQ;
            int bc = NQ - 1 - c, bt = NQ - 1 - t;
            int lo = bc < bt ? bc : bt;
            int hi = bc < bt ? bt : bc;
            for (int q = tid; q < NCPAIR; q += NT) {
                int v = q;
                v = ((v >> lo) << (lo + 1)) | (v & ((1 << lo) - 1));
                v = ((v >> hi) << (hi + 1)) | (v & ((1 << hi) - 1));
                int i0 = v | (1 << bc);        // control = 1, target = 0
                int i1 = i0 | (1 << bt);       // control = 1, target = 1
                float2 t0 = S[i0];
                float2 t1 = S[i1];
                S[i0] = t1;
                S[i1] = t0;
            }
            __syncthreads();
        }
    }

    // <Z_i> = sum_idx |amp|^2 * (1 - 2*bit_i(idx)); deterministic reduction
    float acc[NQ];
#pragma unroll
    for (int w = 0; w < NQ; ++w) acc[w] = 0.f;
    for (int i = tid; i < NSTATE; i += NT) {
        float2 a = S[i];
        float p = a.x * a.x + a.y * a.y;
#pragma unroll
        for (int w = 0; w < NQ; ++w)
            acc[w] += ((i >> (NQ - 1 - w)) & 1) ? -p : p;
    }
#pragma unroll
    for (int w = 0; w < NQ; ++w)
        for (int off = 16; off > 0; off >>= 1)
            acc[w] += __shfl_down(acc[w], off, 32);

    int wave = tid >> 5, lane = tid & 31;
    if (lane == 0)
        for (int w = 0; w < NQ; ++w) wacc[wave][w] = acc[w];
    __syncthreads();
    // write zero-padded row of 16 so the WMMA head can load unconditionally
    if (tid < NQPAD) {
        float e = 0.f;
        if (tid < NQ)
            for (int wv = 0; wv < NT / 32; ++wv) e += wacc[wv][tid];
        expvals[b * NQPAD + tid] = e;
    }
}

// ---------------- kernel 2: WMMA head  logits = E(128x16pad) @ w + b -------
// A 16x4 f32 fragment: lanes 0-15 hold K=0,1 ; lanes 16-31 hold K=2,3.
// B replicated across all N columns -> every D column equals the logits.
__global__ __launch_bounds__(32)
void head_kernel(const float* __restrict__ ev, const float* __restrict__ w_lin,
                 const float* __restrict__ b_lin, float* __restrict__ out) {
    __shared__ float ws[NQPAD];             // zero-padded weight vector
    const int lane  = threadIdx.x;          // one wave32
    const int mtile = blockIdx.x;           // 8 tiles of 16 rows
    const int m     = mtile * 16 + (lane & 15);
    const int khalf = lane >> 4;            // 0: K=0,1  1: K=2,3

    if (lane < NQPAD) ws[lane] = (lane < NQ) ? w_lin[lane] : 0.f;
    __syncthreads();

    v8f c = {};
#pragma unroll
    for (int kc = 0; kc < 4; ++kc) {        // K = 16 (14 real + 2 zero pad)
        int k0 = kc * 4 + khalf * 2;        // even -> 8B-aligned b64 load
        v2f a  = *(const v2f*)(ev + m * NQPAD + k0);
        v2f bv;
        bv.x = ws[k0];
        bv.y = ws[k0 + 1];
        c = __builtin_amdgcn_wmma_f32_16x16x4_f32(false, a, false, bv,
                                                  (short)0, c, false, false);
    }
    // D layout: VGPR r -> lanes 0-15: M=r, lanes 16-31: M=r+8 (all N equal)
    float bias = b_lin[0];
    if ((lane & 15) == 0) {
        int mbase = mtile * 16 + (lane >> 4) * 8;
#pragma unroll
        for (int r = 0; r < 8; ++r) out[mbase + r] = c[r] + bias;
    }
}

// ---------------- launch ---------------------------------------------------
extern "C" void kernel_launch(void* const* d_in, const int* in_sizes, int n_in,
                              void* d_out, int out_size, void* d_ws, size_t ws_size,
                              hipStream_t stream) {
    const float* x     = (const float*)d_in[0];   // (128, 14)
    const float* qw    = (const float*)d_in[1];   // (3, 14, 3)
    const float* w_lin = (const float*)d_in[2];   // (1, 14)
    const float* b_lin = (const float*)d_in[3];   // (1,)
    float* out = (float*)d_out;                   // (128, 1)
    float* expvals = (float*)d_ws;                // (128, 16) padded scratch

    size_t smem = NSTATE * sizeof(float2);        // 128 KB, fits 320 KB LDS/WGP
    qsim_kernel<<<BATCH, NT, smem, stream>>>(x, qw, expvals);
    head_kernel<<<BATCH / 16, 32, 0, stream>>>(expvals, w_lin, b_lin, out);
}